// DeformableAttention_10471130268138
// MI455X (gfx1250) — compile-verified
//
#include <hip/hip_runtime.h>
#include <hip/hip_bf16.h>
#include <stdint.h>

// Problem constants (match reference)
#define Cc   512
#define NHh  8
#define NPp  4
#define HDd  64
#define Hh   20
#define Ww   20
#define Dd   20
#define Nn   8000     // H*W*D
#define Bb   2
#define Mtot 16000    // B*N

typedef __attribute__((ext_vector_type(2))) float v2f;
typedef __attribute__((ext_vector_type(8))) float v8f;
typedef __attribute__((ext_vector_type(4))) unsigned int u32x4;
typedef __attribute__((ext_vector_type(8))) int i32x8;
typedef __attribute__((ext_vector_type(4))) int i32x4;

// ---------------------------------------------------------------------------
// Tensor Data Mover: 2D tile load Global -> LDS (CDNA5 D#, ISA ch.8)
// tile0 = contiguous dim (elements), tile1 = rows, stride = row stride (elems)
// pad codes: interval = 2<<pad_int DWORDs, amount = pad_amt+1 DWORDs
// ---------------------------------------------------------------------------
__device__ __forceinline__ void tdm_load_2d(unsigned lds_off, const void* gptr,
                                            unsigned tile0, unsigned tile1,
                                            unsigned long long stride_elems,
                                            unsigned pad_en, unsigned pad_int,
                                            unsigned pad_amt) {
    unsigned long long ga = (unsigned long long)(uintptr_t)gptr;
    u32x4 g0;
    g0[0] = 1u;                                             // count=1, user mode, no gather
    g0[1] = lds_off;                                        // lds_addr
    g0[2] = (unsigned)(ga & 0xFFFFFFFFu);                   // global_addr lo
    g0[3] = (unsigned)((ga >> 32) & 0x01FFFFFFu) | 0x80000000u;  // global_addr hi | type=2
    i32x8 g1;
    g1[0] = (int)((2u << 16) | (pad_en << 20) | (pad_int << 22) | (pad_amt << 25)); // data_size=4B
    g1[1] = (int)((tile0 & 0xFFFFu) << 16);                 // tensor_dim0 lo16 (== tile0)
    g1[2] = (int)(((tile0 >> 16) & 0xFFFFu) | ((tile1 & 0xFFFFu) << 16)); // dim0 hi | dim1 lo
    g1[3] = (int)(((tile1 >> 16) & 0xFFFFu) | ((tile0 & 0xFFFFu) << 16)); // dim1 hi | tile_dim0
    g1[4] = (int)(tile1 & 0xFFFFu);                         // tile_dim1 ; tile_dim2 = 0
    g1[5] = (int)(stride_elems & 0xFFFFFFFFu);              // tensor_dim0_stride lo32
    g1[6] = (int)((stride_elems >> 32) & 0xFFFFu);          // stride hi16 ; dim1_stride lo = 0
    g1[7] = 0;
    i32x4 z4 = {0, 0, 0, 0};
#if defined(__clang_major__) && (__clang_major__ >= 23)
    i32x8 z8 = {0, 0, 0, 0, 0, 0, 0, 0};
    __builtin_amdgcn_tensor_load_to_lds(g0, g1, z4, z4, z8, 0);
#else
    __builtin_amdgcn_tensor_load_to_lds(g0, g1, z4, z4, 0);
#endif
}

// ---------------------------------------------------------------------------
// LayerNorm + transpose: f_query (B,C,N) -> q (M,C) row-major.
// Block handles 32 consecutive tokens; lanes ride the N axis so every global
// read/write is a contiguous 128B line; transpose goes through padded LDS.
// ---------------------------------------------------------------------------
__global__ __launch_bounds__(256) void ln_kernel(const float* __restrict__ fq,
                                                 const float* __restrict__ g,
                                                 const float* __restrict__ bta,
                                                 float* __restrict__ q) {
    __shared__ float red[8][32], red2[8][32];
    __shared__ float mu_s[32], rs_s[32];
    __shared__ float tile[32][33];

    int t = threadIdx.x;
    int lane = t & 31;                 // token offset (pass1/2 reads), channel (writes)
    int grp  = t >> 5;                 // 0..7
    int m0 = blockIdx.x * 32;          // 32 tokens, never crosses batch (8000 % 32 == 0)
    int b  = m0 / Nn;
    int n0 = m0 - b * Nn;
    const float* fb = fq + (long)b * Cc * Nn + n0;

    // pass 1: per-token mean/var (coalesced along N)
    float s = 0.f, ss = 0.f;
    for (int c = grp; c < Cc; c += 8) {
        float x = fb[(long)c * Nn + lane];
        s += x; ss += x * x;
    }
    red[grp][lane] = s; red2[grp][lane] = ss;
    __syncthreads();
    if (t < 32) {
        float S = 0.f, SS = 0.f;
#pragma unroll
        for (int j = 0; j < 8; ++j) { S += red[j][t]; SS += red2[j][t]; }
        float mu = S * (1.f / Cc);
        float var = SS * (1.f / Cc) - mu * mu;
        mu_s[t] = mu; rs_s[t] = rsqrtf(var + 1e-5f);
    }
    __syncthreads();

    // pass 2: normalize + LDS transpose, 32-channel chunks
    for (int cc = 0; cc < Cc; cc += 32) {
#pragma unroll
        for (int i = 0; i < 4; ++i) {
            int c = cc + grp * 4 + i;
            float x = fb[(long)c * Nn + lane];
            tile[grp * 4 + i][lane] = (x - mu_s[lane]) * rs_s[lane] * g[c] + bta[c];
        }
        __syncthreads();
#pragma unroll
        for (int i = 0; i < 4; ++i) {
            int tok = grp + i * 8;
            q[(long)(m0 + tok) * Cc + cc + lane] = tile[lane][tok];
        }
        __syncthreads();
    }
}

// ---------------------------------------------------------------------------
// 512x512 transpose (weights -> (N,K) so B fragments are K-contiguous b64)
// ---------------------------------------------------------------------------
__global__ __launch_bounds__(256) void transpose512(const float* __restrict__ W,
                                                    float* __restrict__ Wt) {
    __shared__ float tl[32][33];
    int bx = blockIdx.x & 15, by = blockIdx.x >> 4;
    int lane = threadIdx.x & 31, grp = threadIdx.x >> 5;
#pragma unroll
    for (int i = 0; i < 4; ++i)
        tl[grp * 4 + i][lane] = W[(by * 32 + grp * 4 + i) * 512 + bx * 32 + lane];
    __syncthreads();
#pragma unroll
    for (int i = 0; i < 4; ++i)
        Wt[(bx * 32 + grp * 4 + i) * 512 + by * 32 + lane] = tl[lane][grp * 4 + i];
}

// ---------------------------------------------------------------------------
// Big LDS-staged WMMA GEMM:  D(M,Nout) = A(M,K) * Wt(Nout,K)^T + bias
// (M % 128 == 0, Nout % 64 == 0, K % 32 == 0)
// Block = 256 thr = 8 waves (2Mx4N); block tile 128x64; wave tile 64x16.
// K staged 32 at a time via double-buffered Tensor Data Mover; both A and B
// LDS tiles are [row][k] with stride 34 -> all fragment reads are b64,
// bank-conflict-free.
//   EPI: 0=none, 1=relu, 3=store transposed to (B, Nout, batchN)
// ---------------------------------------------------------------------------
template <int EPI>
__global__ __launch_bounds__(256) void gemm_wmma_lds(const float* __restrict__ A,
                                                     const float* __restrict__ Wt,
                                                     const float* __restrict__ bias,
                                                     float* __restrict__ Dm,
                                                     int M, int K, int Nout,
                                                     int batchN, int tilesM) {
    __shared__ float Alds[2][128 * 34];   // 2 x 17408B
    __shared__ float Blds[2][64 * 34];    // 2 x 8704B

    int wid = threadIdx.x >> 5, lane = threadIdx.x & 31;
    int tm = blockIdx.x % tilesM, tn = blockIdx.x / tilesM;
    int m0 = tm << 7;
    int n0c = tn << 6;
    int wm = wid >> 2, wn = wid & 3;
    int lrow = lane & 15;
    int kq = (lane >> 4) << 1;
    int colg = n0c + (wn << 4) + lrow;

    unsigned ldsA[2] = { (unsigned)(uintptr_t)(void*)&Alds[0][0],
                         (unsigned)(uintptr_t)(void*)&Alds[1][0] };
    unsigned ldsB[2] = { (unsigned)(uintptr_t)(void*)&Blds[0][0],
                         (unsigned)(uintptr_t)(void*)&Blds[1][0] };

    const int NS = K >> 5;
    // prologue: stage 0 into buffer 0
    if (wid == 0) {
        tdm_load_2d(ldsA[0], A + (long)m0 * K, 32u, 128u, (unsigned long long)K, 1u, 4u, 1u);
        tdm_load_2d(ldsB[0], Wt + (long)n0c * K, 32u, 64u, (unsigned long long)K, 1u, 4u, 1u);
    }

    v8f acc[4] = {};
    for (int s = 0; s < NS; ++s) {
        int cur = s & 1;
        if (wid == 0) __builtin_amdgcn_s_wait_tensorcnt(0);
        __syncthreads();                              // publish buffer `cur`
        if (wid == 0 && s + 1 < NS) {                 // prefetch next stage into other buffer
            int ks = (s + 1) << 5;
            tdm_load_2d(ldsA[cur ^ 1], A + (long)m0 * K + ks, 32u, 128u,
                        (unsigned long long)K, 1u, 4u, 1u);
            tdm_load_2d(ldsB[cur ^ 1], Wt + (long)n0c * K + ks, 32u, 64u,
                        (unsigned long long)K, 1u, 4u, 1u);
        }
        const float* Aptr = &Alds[cur][0];
        const float* Bptr = &Blds[cur][0];
#pragma unroll
        for (int kk = 0; kk < 32; kk += 4) {
            int k0 = kk + kq;
            v2f bf = *(const v2f*)&Bptr[((wn << 4) + lrow) * 34 + k0];
#pragma unroll
            for (int s4 = 0; s4 < 4; ++s4) {
                int rl = (wm << 6) + (s4 << 4) + lrow;
                v2f af = *(const v2f*)&Aptr[rl * 34 + k0];
                acc[s4] = __builtin_amdgcn_wmma_f32_16x16x4_f32(
                    false, af, false, bf, (short)0, acc[s4], false, false);
            }
        }
        // next iteration's barrier fences reads of `cur` before it is refilled
    }

    float bv = bias[colg];
#pragma unroll
    for (int s4 = 0; s4 < 4; ++s4) {
#pragma unroll
        for (int e = 0; e < 8; ++e) {
            int m = m0 + (wm << 6) + (s4 << 4) + ((lane >> 4) << 3) + e;
            float v = acc[s4][e] + bv;
            if (EPI == 1) v = fmaxf(v, 0.f);
            if (EPI == 3) {
                int b2 = m / batchN, n2 = m - b2 * batchN;
                Dm[((long)b2 * Nout + colg) * batchN + n2] = v;
            } else {
                Dm[(long)m * Nout + colg] = v;
            }
        }
    }
}

// ---------------------------------------------------------------------------
// Small direct WMMA GEMM for narrow Nout (96 / 32). A row-major (M,K).
// EPI: 0=none, 2=clip(+-3). Wave tile 64x16.
// ---------------------------------------------------------------------------
template <int EPI>
__global__ __launch_bounds__(256) void gemm_wmma_small(const float* __restrict__ A,
                                                       const float* __restrict__ Wm,
                                                       const float* __restrict__ bias,
                                                       float* __restrict__ Dm,
                                                       int M, int K, int Nout) {
    int wid = threadIdx.x >> 5, lane = threadIdx.x & 31;
    int tilesM = M >> 6, tilesN = Nout >> 4;
    int t = blockIdx.x * 8 + wid;
    if (t >= tilesM * tilesN) return;            // whole-wave exit: EXEC stays all-1s
    int tm = t % tilesM, tn = t / tilesM;
    int row0 = tm << 6;
    int col  = (tn << 4) + (lane & 15);
    int kq   = (lane >> 4) << 1;
    int lrow = lane & 15;

    v8f acc[4] = {};
    for (int k = 0; k < K; k += 4) {
        int k0 = k + kq;
        v2f bf;
        bf.x = Wm[k0 * Nout + col];
        bf.y = Wm[(k0 + 1) * Nout + col];
#pragma unroll
        for (int s4 = 0; s4 < 4; ++s4) {
            v2f af = *(const v2f*)(A + (long)(row0 + (s4 << 4) + lrow) * K + k0);
            acc[s4] = __builtin_amdgcn_wmma_f32_16x16x4_f32(
                false, af, false, bf, (short)0, acc[s4], false, false);
        }
    }
    float bv = bias[col];
#pragma unroll
    for (int s4 = 0; s4 < 4; ++s4) {
#pragma unroll
        for (int e = 0; e < 8; ++e) {
            int m = row0 + (s4 << 4) + ((lane >> 4) << 3) + e;
            float v = acc[s4][e] + bv;
            if (EPI == 2) v = fminf(fmaxf(v, -3.f), 3.f);
            Dm[(long)m * Nout + col] = v;
        }
    }
}

// ---------------------------------------------------------------------------
// Permute f_kv (B,C,H,W,D) -> feats (B,NH,D,H,W,HD)  (channels-last gather)
// ---------------------------------------------------------------------------
__global__ __launch_bounds__(256) void permute_kernel(const float* __restrict__ fkv,
                                                      float* __restrict__ feats) {
    int i = blockIdx.x * 256 + threadIdx.x;
    if (i >= Bb * Cc * Nn) return;
    int hd = i & 63;
    int x  = (i >> 6) % Ww;
    int y  = (i / (64 * Ww)) % Hh;
    int z  = (i / (64 * Ww * Hh)) % Dd;
    int h  = (i / (64 * Ww * Hh * Dd)) % NHh;
    int b  =  i / (64 * Ww * Hh * Dd * NHh);
    int c  = h * HDd + hd;
    feats[i] = fkv[(((long)(b * Cc + c) * Hh + y) * Ww + x) * Dd + z];
}

// ---------------------------------------------------------------------------
// Softmax over NP=4 per (token, head), in place on (M, NH*NP)
// ---------------------------------------------------------------------------
__global__ __launch_bounds__(256) void softmax_kernel(float* __restrict__ attn) {
    int t = blockIdx.x * 256 + threadIdx.x;
    if (t >= Mtot * NHh) return;
    float* p = attn + (long)t * 4;
    float a0 = p[0], a1 = p[1], a2 = p[2], a3 = p[3];
    float mx = fmaxf(fmaxf(a0, a1), fmaxf(a2, a3));
    float e0 = expf(a0 - mx), e1 = expf(a1 - mx), e2 = expf(a2 - mx), e3 = expf(a3 - mx);
    float r = 1.f / (e0 + e1 + e2 + e3);
    p[0] = e0 * r; p[1] = e1 * r; p[2] = e2 * r; p[3] = e3 * r;
}

// ---------------------------------------------------------------------------
// Trilinear sampling + attention combine. One wave per (token, head).
// ---------------------------------------------------------------------------
__global__ __launch_bounds__(256) void sample_kernel(const float* __restrict__ feats,
                                                     const float* __restrict__ off,
                                                     const float* __restrict__ attn,
                                                     float* __restrict__ val) {
    int wid = threadIdx.x >> 5, lane = threadIdx.x & 31;
    int idx = blockIdx.x * 8 + wid;
    if (idx >= Mtot * NHh) return;
    int m = idx >> 3;
    int h = idx & 7;
    int b = m / Nn, n = m - b * Nn;
    int ny = n / (Ww * Dd);
    int nx = (n / Dd) % Ww;
    int nz = n % Dd;
    const float* fb = feats + (long)(b * NHh + h) * (Dd * Hh * Ww * HDd);
    const float* op = off + (long)idx * (NPp * 3);
    const float* ap = attn + (long)idx * NPp;

    float acc0 = 0.f, acc1 = 0.f;
#pragma unroll
    for (int p = 0; p < NPp; ++p) {
        // H==W==D==20 makes the reference's cross-normalization collapse to identity
        float ix = fminf(fmaxf((float)ny + op[p * 3 + 0], 0.f), (float)(Ww - 1));
        float iy = fminf(fmaxf((float)nx + op[p * 3 + 1], 0.f), (float)(Hh - 1));
        float iz = fminf(fmaxf((float)nz + op[p * 3 + 2], 0.f), (float)(Dd - 1));
        float x0f = floorf(ix), y0f = floorf(iy), z0f = floorf(iz);
        float fx = ix - x0f, fy = iy - y0f, fz = iz - z0f;
        int x0 = (int)x0f, y0 = (int)y0f, z0 = (int)z0f;
        int x1 = min(x0 + 1, Ww - 1), y1 = min(y0 + 1, Hh - 1), z1 = min(z0 + 1, Dd - 1);
        float gx = 1.f - fx, gy = 1.f - fy, gz = 1.f - fz;
        float a = ap[p];

        const float* c000 = fb + ((z0 * Hh + y0) * Ww + x0) * HDd;
        const float* c001 = fb + ((z0 * Hh + y0) * Ww + x1) * HDd;
        const float* c010 = fb + ((z0 * Hh + y1) * Ww + x0) * HDd;
        const float* c011 = fb + ((z0 * Hh + y1) * Ww + x1) * HDd;
        const float* c100 = fb + ((z1 * Hh + y0) * Ww + x0) * HDd;
        const float* c101 = fb + ((z1 * Hh + y0) * Ww + x1) * HDd;
        const float* c110 = fb + ((z1 * Hh + y1) * Ww + x0) * HDd;
        const float* c111 = fb + ((z1 * Hh + y1) * Ww + x1) * HDd;

        float w000 = gz * gy * gx, w001 = gz * gy * fx;
        float w010 = gz * fy * gx, w011 = gz * fy * fx;
        float w100 = fz * gy * gx, w101 = fz * gy * fx;
        float w110 = fz * fy * gx, w111 = fz * fy * fx;

        float s0 = w000 * c000[lane] + w001 * c001[lane] + w010 * c010[lane] + w011 * c011[lane]
                 + w100 * c100[lane] + w101 * c101[lane] + w110 * c110[lane] + w111 * c111[lane];
        float s1 = w000 * c000[lane + 32] + w001 * c001[lane + 32] + w010 * c010[lane + 32] + w011 * c011[lane + 32]
                 + w100 * c100[lane + 32] + w101 * c101[lane + 32] + w110 * c110[lane + 32] + w111 * c111[lane + 32];
        acc0 += a * s0;
        acc1 += a * s1;
    }
    float* vp = val + (long)m * Cc + h * HDd;
    vp[lane] = acc0;
    vp[lane + 32] = acc1;
}

// ---------------------------------------------------------------------------
// Host-side launch sequence
// ---------------------------------------------------------------------------
static inline int gemm_small_blocks(int M, int Nout) {
    int tiles = (M >> 6) * (Nout >> 4);
    return (tiles + 7) / 8;
}

extern "C" void kernel_launch(void* const* d_in, const int* in_sizes, int n_in,
                              void* d_out, int out_size, void* d_ws, size_t ws_size,
                              hipStream_t stream) {
    const float* f_query = (const float*)d_in[0];
    const float* f_kv    = (const float*)d_in[1];
    const float* ln_g    = (const float*)d_in[2];
    const float* ln_b    = (const float*)d_in[3];
    const float* Wq      = (const float*)d_in[4];
    const float* bq      = (const float*)d_in[5];
    const float* Wo1     = (const float*)d_in[6];
    const float* bo1     = (const float*)d_in[7];
    const float* Wo2     = (const float*)d_in[8];
    const float* bo2     = (const float*)d_in[9];
    const float* Wa      = (const float*)d_in[10];
    const float* ba      = (const float*)d_in[11];
    const float* Wout    = (const float*)d_in[12];
    const float* bout    = (const float*)d_in[13];
    float* out = (float*)d_out;

    const size_t SZ = (size_t)Mtot * Cc;        // 8,192,000 floats
    float* ws0  = (float*)d_ws;                 // q (M,C) -> feats
    float* ws1  = ws0 + SZ;                     // Q -> val
    float* ws2  = ws1 + SZ;                     // hidden
    float* offb = ws2 + SZ;                     // (M, 96)
    float* attb = offb + (size_t)Mtot * (NHh * NPp * 3);  // (M, 32)
    float* wt0  = attb + (size_t)Mtot * (NHh * NPp);      // Wq^T   (512x512)
    float* wt1  = wt0 + (size_t)Cc * Cc;                  // Wo1^T
    float* wt2  = wt1 + (size_t)Cc * Cc;                  // Wout^T

    const int tilesM = Mtot >> 7;               // 125
    const int bigGrid = tilesM * (Cc >> 6);     // 125 * 8 = 1000

    // 0. transpose the three big weight matrices to (N,K)
    transpose512<<<256, 256, 0, stream>>>(Wq, wt0);
    transpose512<<<256, 256, 0, stream>>>(Wo1, wt1);
    transpose512<<<256, 256, 0, stream>>>(Wout, wt2);

    // 1. LayerNorm + transpose: q (M,C) row-major
    ln_kernel<<<Mtot / 32, 256, 0, stream>>>(f_query, ln_g, ln_b, ws0);

    // 2. Q = q @ Wq + bq
    gemm_wmma_lds<0><<<bigGrid, 256, 0, stream>>>(
        ws0, wt0, bq, ws1, Mtot, Cc, Cc, Nn, tilesM);

    // 3. feats channels-last (reuses ws0 after q is consumed)
    permute_kernel<<<(Bb * Cc * Nn + 255) / 256, 256, 0, stream>>>(f_kv, ws0);

    // 4. hidden = relu(Q @ Wo1 + bo1)
    gemm_wmma_lds<1><<<bigGrid, 256, 0, stream>>>(
        ws1, wt1, bo1, ws2, Mtot, Cc, Cc, Nn, tilesM);

    // 5. off = clip(hidden @ Wo2 + bo2, +-3)   (M, 96)
    gemm_wmma_small<2><<<gemm_small_blocks(Mtot, NHh * NPp * 3), 256, 0, stream>>>(
        ws2, Wo2, bo2, offb, Mtot, Cc, NHh * NPp * 3);

    // 6. attn_raw = Q @ Wa + ba               (M, 32)
    gemm_wmma_small<0><<<gemm_small_blocks(Mtot, NHh * NPp), 256, 0, stream>>>(
        ws1, Wa, ba, attb, Mtot, Cc, NHh * NPp);

    // 7. softmax over NP in place
    softmax_kernel<<<(Mtot * NHh + 255) / 256, 256, 0, stream>>>(attb);

    // 8. trilinear sample + combine -> val (M, C)   (reuses ws1 after Q consumed)
    sample_kernel<<<Mtot * NHh / 8, 256, 0, stream>>>(ws0, offb, attb, ws1);

    // 9. out = val @ Wout + bout, stored transposed to (B,C,N)
    gemm_wmma_lds<3><<<bigGrid, 256, 0, stream>>>(
        ws1, wt2, bout, out, Mtot, Cc, Cc, Nn, tilesM);
}